// PointNetSetAbstractionMsg_59210419143151
// MI455X (gfx1250) — compile-verified
//
#include <hip/hip_runtime.h>
#include <stdint.h>

typedef __attribute__((ext_vector_type(2))) float v2f;
typedef __attribute__((ext_vector_type(4))) float v4f;
typedef __attribute__((ext_vector_type(8))) float v8f;

#define B_    8
#define N_    4096
#define D_    32
#define S_    1024
#define CTOT_ 640
#define CINP_ 36          // 3 + D_ = 35, padded to multiple of 4 for clean WMMA K-loop

// LDS row stride for the staged B matrix: ≡ 4 (mod 64) words so the 16 active
// lanes of a b64 read hit banks 0,4,8,...,60 (conflict-free).
__host__ __device__ constexpr int lds_stride(int cinpad) {
  return ((cinpad + 63) / 64) * 64 + 4;
}

// ---------------------------------------------------------------------------
// Farthest point sampling: one block per batch, 1024 serial argmax rounds.
// Tie-break = smallest index (matches jnp.argmax first-occurrence).
// ---------------------------------------------------------------------------
__global__ void fps_kernel(const float* __restrict__ xyz, int* __restrict__ fps_idx) {
  const int b = blockIdx.x;
  const int tid = threadIdx.x;
  const float* xb = xyz + (size_t)b * N_ * 3;
  float dist[N_ / 256];
#pragma unroll
  for (int j = 0; j < N_ / 256; ++j) dist[j] = 1e10f;
  __shared__ float sval[256];
  __shared__ int   sidx[256];
  int far = 0;
  for (int it = 0; it < S_; ++it) {
    if (tid == 0) fps_idx[b * S_ + it] = far;
    const float cx = xb[far * 3 + 0];
    const float cy = xb[far * 3 + 1];
    const float cz = xb[far * 3 + 2];
    float best = -1.0f;
    int   bi = 0x7fffffff;
#pragma unroll
    for (int j = 0; j < N_ / 256; ++j) {
      const int p = tid + j * 256;
      const float dx = xb[p * 3 + 0] - cx;
      const float dy = xb[p * 3 + 1] - cy;
      const float dz = xb[p * 3 + 2] - cz;
      float d = dx * dx + dy * dy + dz * dz;
      d = fminf(dist[j], d);
      dist[j] = d;
      if (d > best) { best = d; bi = p; }   // p ascends within thread -> first max kept
    }
    sval[tid] = best; sidx[tid] = bi;
    __syncthreads();
    for (int s = 128; s > 0; s >>= 1) {
      if (tid < s) {
        const float v2 = sval[tid + s]; const int i2 = sidx[tid + s];
        if (v2 > sval[tid] || (v2 == sval[tid] && i2 < sidx[tid])) { sval[tid] = v2; sidx[tid] = i2; }
      }
      __syncthreads();
    }
    far = sidx[0];
    __syncthreads();
  }
}

__global__ void gather_newxyz_kernel(const float* __restrict__ xyz, const int* __restrict__ fps_idx,
                                     float* __restrict__ new_xyz, float* __restrict__ out_xyz) {
  const int t = blockIdx.x * blockDim.x + threadIdx.x;
  if (t >= B_ * S_) return;
  const int b = t / S_;
  const int p = fps_idx[t];
  const float* src = xyz + ((size_t)b * N_ + p) * 3;
#pragma unroll
  for (int c = 0; c < 3; ++c) { const float v = src[c]; new_xyz[t * 3 + c] = v; out_xyz[t * 3 + c] = v; }
}

// ---------------------------------------------------------------------------
// Ball query: first K indices (ascending) with d<=r^2, padded with first hit.
// Chunked with software prefetch of the xyz stream (global_prefetch_b8).
// ---------------------------------------------------------------------------
__global__ void ball_query_kernel(const float* __restrict__ xyz, const float* __restrict__ new_xyz,
                                  int* __restrict__ idx, float r2, int K) {
  const int t = blockIdx.x * blockDim.x + threadIdx.x;
  if (t >= B_ * S_) return;
  const int b = t / S_;
  const float* xb = xyz + (size_t)b * N_ * 3;
  const float cx = new_xyz[t * 3 + 0];
  const float cy = new_xyz[t * 3 + 1];
  const float cz = new_xyz[t * 3 + 2];
  int* out = idx + (size_t)t * K;
  int cnt = 0, first = 0;
  for (int p0 = 0; p0 < N_ && cnt < K; p0 += 128) {
    if (p0 + 128 < N_) __builtin_prefetch(xb + (size_t)(p0 + 128) * 3, 0, 1);
    for (int p = p0; p < p0 + 128; ++p) {
      const float dx = xb[p * 3 + 0] - cx;
      const float dy = xb[p * 3 + 1] - cy;
      const float dz = xb[p * 3 + 2] - cz;
      const float d = dx * dx + dy * dy + dz * dz;
      if (d <= r2) {
        if (cnt == 0) first = p;
        out[cnt++] = p;
        if (cnt == K) break;
      }
    }
  }
  for (int k = cnt; k < K; ++k) out[k] = first;
}

// feat[m, 0:3] = xyz[idx]-center ; feat[m, 3:35] = points[idx] ; feat[m,35] = 0
// Row stride CINP_ (36) so the GEMM K-loop needs no bounds guards.
__global__ void build_feat_kernel(const float* __restrict__ xyz, const float* __restrict__ points,
                                  const float* __restrict__ new_xyz, const int* __restrict__ idx,
                                  float* __restrict__ feat, int K) {
  const size_t total = (size_t)B_ * S_ * K;
  const size_t t = (size_t)blockIdx.x * blockDim.x + threadIdx.x;
  if (t >= total) return;
  const int s_flat = (int)(t / K);      // b*S + s
  const int b = s_flat / S_;
  const int p = idx[t];
  float* f = feat + t * CINP_;
  const float* src = xyz + ((size_t)b * N_ + p) * 3;
  const float* cen = new_xyz + (size_t)s_flat * 3;
  f[0] = src[0] - cen[0];
  f[1] = src[1] - cen[1];
  f[2] = src[2] - cen[2];
  const float* pt = points + ((size_t)b * N_ + p) * D_;
#pragma unroll
  for (int c = 0; c < D_; ++c) f[3 + c] = pt[c];
  f[35] = 0.0f;                          // zero pad column
}

// Transpose + zero-pad weights into the bank-conflict-free LDS layout:
// Wt[col*ldw + k] = (k<Cin) ? W[k*Cout+col] : 0   (pads zeroed)
__global__ void prep_weight_kernel(const float* __restrict__ W, float* __restrict__ Wt,
                                   int Cin, int ldw, int Cout) {
  const int t = blockIdx.x * blockDim.x + threadIdx.x;
  if (t >= Cout * ldw) return;
  const int k = t % ldw;
  const int col = t / ldw;
  Wt[t] = (k < Cin) ? W[(size_t)k * Cout + col] : 0.0f;
}

// ---------------------------------------------------------------------------
// GEMM Y = X*W + bias via V_WMMA_F32_16X16X4_F32.
// - One wave owns a 16-row strip and ALL Cout/16 N-tiles (X read exactly once).
// - Wt staged once per block into LDS (8 waves share it): B feed = ds_load_b64,
//   8x less L2 traffic than per-wave global reads.
// - A lanes 0-15: K=k0,k0+1 of row M=lane; lanes 16-31: K=k0+2,k0+3. B mirrored.
// - All strides compile-time; loads unconditional & 8B aligned -> EXEC all-1s.
// ---------------------------------------------------------------------------
template <int CINPAD, int COUT>
__global__ void wmma_gemm_bias_kernel(const float* __restrict__ X, const float* __restrict__ Wt,
                                      const float* __restrict__ bias, float* __restrict__ Y,
                                      int M) {
  constexpr int NT  = COUT / 16;
  constexpr int LDW = lds_stride(CINPAD);
  __shared__ float sW[COUT * LDW];
  const int tid = threadIdx.x;

  // cooperative fill (prep kernel already produced this exact layout)
  for (int i = tid; i < (COUT * LDW) / 4; i += 256)
    reinterpret_cast<v4f*>(sW)[i] = reinterpret_cast<const v4f*>(Wt)[i];
  __syncthreads();

  const int lane  = tid & 31;
  const int mtile = blockIdx.x * 8 + (tid >> 5);   // exact grid: M/16 waves total
  const int l16   = lane & 15;
  const int hi    = lane >> 4;                     // 0 or 1
  const int koff  = hi * 2;
  const float* Xrow = X + (size_t)(mtile * 16 + l16) * CINPAD;

  v8f acc[NT];
#pragma unroll
  for (int nt = 0; nt < NT; ++nt) acc[nt] = {};

#pragma unroll 2
  for (int k0 = 0; k0 < CINPAD; k0 += 4) {
    const int ka = k0 + koff;
    const v2f a = *reinterpret_cast<const v2f*>(Xrow + ka);
#pragma unroll
    for (int nt = 0; nt < NT; ++nt) {
      const v2f bm = *reinterpret_cast<const v2f*>(&sW[(nt * 16 + l16) * LDW + ka]);
      acc[nt] = __builtin_amdgcn_wmma_f32_16x16x4_f32(false, a, false, bm, (short)0,
                                                      acc[nt], false, false);
    }
  }

  const int rbase = mtile * 16 + hi * 8;
#pragma unroll
  for (int nt = 0; nt < NT; ++nt) {
    const int col = nt * 16 + l16;
    const float bv = bias[col];
#pragma unroll
    for (int j = 0; j < 8; ++j)
      Y[(size_t)(rbase + j) * COUT + col] = acc[nt][j] + bv;
  }
  (void)M;
}

// ---------------------------------------------------------------------------
// Deterministic global batchnorm stats. Block handles 1024 rows; since C
// divides 256, each thread only touches channel (tid % C); partials combined
// in a fixed order, then a single-block finalize in fixed order.
// ---------------------------------------------------------------------------
__global__ void stats_partial_kernel(const float* __restrict__ Y, float* __restrict__ partial, int C) {
  const int blk = blockIdx.x;
  const int tid = threadIdx.x;
  const float* Yb = Y + (size_t)blk * 1024 * C;
  float s = 0.f, ss = 0.f;
  const int total = 1024 * C;
  for (int i = tid; i < total; i += 256) {
    const float v = Yb[i];
    s += v; ss += v * v;
  }
  __shared__ float sh[256], shh[256];
  sh[tid] = s; shh[tid] = ss;
  __syncthreads();
  if (tid < C) {
    float a = 0.f, aa = 0.f;
    for (int u = tid; u < 256; u += C) { a += sh[u]; aa += shh[u]; }
    partial[(size_t)blk * 2 * C + tid] = a;
    partial[(size_t)blk * 2 * C + C + tid] = aa;
  }
}

__global__ void stats_finalize_kernel(const float* __restrict__ partial, const float* __restrict__ g,
                                      const float* __restrict__ be, float* __restrict__ scale,
                                      float* __restrict__ shift, int nblocks, int C, float invM) {
  const int c = blockIdx.x * blockDim.x + threadIdx.x;
  if (c >= C) return;
  float s = 0.f, ss = 0.f;
  for (int b = 0; b < nblocks; ++b) {
    s  += partial[(size_t)b * 2 * C + c];
    ss += partial[(size_t)b * 2 * C + C + c];
  }
  const float mean = s * invM;
  const float var  = ss * invM - mean * mean;
  const float sc   = g[c] * rsqrtf(var + 1e-5f);
  scale[c] = sc;
  shift[c] = be[c] - mean * sc;
}

__global__ void norm_relu_kernel(float* __restrict__ Y, const float* __restrict__ scale,
                                 const float* __restrict__ shift, size_t total, int C) {
  const size_t stride = (size_t)gridDim.x * blockDim.x;
  for (size_t i = (size_t)blockIdx.x * blockDim.x + threadIdx.x; i < total; i += stride) {
    const int c = (int)(i % C);
    const float v = Y[i] * scale[c] + shift[c];
    Y[i] = v > 0.f ? v : 0.f;
  }
}

// max over K, write transposed (B, CTOT, S) slab at channel offset.
__global__ void maxpool_out_kernel(const float* __restrict__ Y, float* __restrict__ out_points,
                                   int K, int C, int chanOff) {
  const size_t total = (size_t)B_ * S_ * C;
  const size_t t = (size_t)blockIdx.x * blockDim.x + threadIdx.x;
  if (t >= total) return;
  const int c = (int)(t % C);
  const int s = (int)((t / C) % S_);
  const int b = (int)(t / ((size_t)C * S_));
  const float* y = Y + ((size_t)(b * S_ + s) * K) * C + c;
  float m = y[0];
  for (int k = 1; k < K; ++k) m = fmaxf(m, y[(size_t)k * C]);
  out_points[(size_t)b * CTOT_ * S_ + (size_t)(chanOff + c) * S_ + s] = m;
}

// ---------------------------------------------------------------------------
static inline int cdiv(long long a, long long b) { return (int)((a + b - 1) / b); }

static void launch_gemm(int CinPad, int Cout, const float* X, const float* Wt, const float* bias,
                        float* Y, int M, hipStream_t stream) {
  const int blocks = M / 128;            // M/16 waves, 8 waves per 256-thread block (exact)
  if (CinPad == 36 && Cout == 64)
    wmma_gemm_bias_kernel<36, 64><<<blocks, 256, 0, stream>>>(X, Wt, bias, Y, M);
  else if (CinPad == 36 && Cout == 128)
    wmma_gemm_bias_kernel<36, 128><<<blocks, 256, 0, stream>>>(X, Wt, bias, Y, M);
  else if (CinPad == 64 && Cout == 128)
    wmma_gemm_bias_kernel<64, 128><<<blocks, 256, 0, stream>>>(X, Wt, bias, Y, M);
  else
    wmma_gemm_bias_kernel<128, 256><<<blocks, 256, 0, stream>>>(X, Wt, bias, Y, M);
}

extern "C" void kernel_launch(void* const* d_in, const int* in_sizes, int n_in,
                              void* d_out, int out_size, void* d_ws, size_t ws_size,
                              hipStream_t stream) {
  (void)in_sizes; (void)n_in; (void)out_size; (void)ws_size;
  const float* xyz    = (const float*)d_in[0];
  const float* points = (const float*)d_in[1];

  float* out        = (float*)d_out;
  float* out_xyz    = out;                          // (8,1024,3)
  float* out_points = out + (size_t)B_ * S_ * 3;    // (8,640,1024)

  // workspace carve-up (needs ~808 MB)
  char* ws = (char*)d_ws;
  size_t off = 0;
  auto carve = [&](size_t bytes) -> void* {
    void* p = ws + off;
    off += (bytes + 255) & ~(size_t)255;
    return p;
  };
  int*   fps_idx  = (int*)  carve((size_t)B_ * S_ * 4);
  float* new_xyz  = (float*)carve((size_t)B_ * S_ * 3 * 4);
  int*   idxbuf   = (int*)  carve((size_t)B_ * S_ * 64 * 4);
  float* partial  = (float*)carve((size_t)512 * 2 * 256 * 4);
  float* scalebuf = (float*)carve(256 * 4);
  float* shiftbuf = (float*)carve(256 * 4);
  float* wtbuf    = (float*)carve((size_t)256 * 132 * 4);           // padded/transposed weights
  float* bufB     = (float*)carve((size_t)B_ * S_ * 64 * 128 * 4);  // layer-1 outs
  float* bufA     = (float*)carve((size_t)B_ * S_ * 64 * 256 * 4);  // feat / layer-2 outs

  fps_kernel<<<B_, 256, 0, stream>>>(xyz, fps_idx);
  gather_newxyz_kernel<<<cdiv(B_ * S_, 256), 256, 0, stream>>>(xyz, fps_idx, new_xyz, out_xyz);

  const float radii[3] = {0.1f, 0.2f, 0.4f};
  const int   nsamp[3] = {16, 32, 64};
  const int   c1[3]    = {64, 128, 128};
  const int   c2[3]    = {128, 256, 256};
  const int   choff[3] = {0, 128, 384};

  for (int i = 0; i < 3; ++i) {
    const int base = 2 + i * 8;
    const float* w0 = (const float*)d_in[base + 0];
    const float* b0 = (const float*)d_in[base + 1];
    const float* g0 = (const float*)d_in[base + 2];
    const float* e0 = (const float*)d_in[base + 3];
    const float* w1 = (const float*)d_in[base + 4];
    const float* b1 = (const float*)d_in[base + 5];
    const float* g1 = (const float*)d_in[base + 6];
    const float* e1 = (const float*)d_in[base + 7];
    const int K = nsamp[i];
    const int M = B_ * S_ * K;

    ball_query_kernel<<<cdiv(B_ * S_, 256), 256, 0, stream>>>(xyz, new_xyz, idxbuf,
                                                              radii[i] * radii[i], K);
    build_feat_kernel<<<cdiv(M, 256), 256, 0, stream>>>(xyz, points, new_xyz, idxbuf, bufA, K);

    { // layer 1: bufA (M x CINP_) -> bufB (M x c1)
      const int Cin = 3 + D_, CinPad = CINP_, Cout = c1[i];
      const int ldw = lds_stride(CinPad);
      prep_weight_kernel<<<cdiv((long long)Cout * ldw, 256), 256, 0, stream>>>(
          w0, wtbuf, Cin, ldw, Cout);
      launch_gemm(CinPad, Cout, bufA, wtbuf, b0, bufB, M, stream);
      const int nblk = M / 1024;
      stats_partial_kernel<<<nblk, 256, 0, stream>>>(bufB, partial, Cout);
      stats_finalize_kernel<<<1, 256, 0, stream>>>(partial, g0, e0, scalebuf, shiftbuf,
                                                   nblk, Cout, 1.0f / (float)M);
      norm_relu_kernel<<<2048, 256, 0, stream>>>(bufB, scalebuf, shiftbuf, (size_t)M * Cout, Cout);
    }
    { // layer 2: bufB (M x c1) -> bufA (M x c2)
      const int Cin = c1[i], CinPad = c1[i], Cout = c2[i];
      const int ldw = lds_stride(CinPad);
      prep_weight_kernel<<<cdiv((long long)Cout * ldw, 256), 256, 0, stream>>>(
          w1, wtbuf, Cin, ldw, Cout);
      launch_gemm(CinPad, Cout, bufB, wtbuf, b1, bufA, M, stream);
      const int nblk = M / 1024;
      stats_partial_kernel<<<nblk, 256, 0, stream>>>(bufA, partial, Cout);
      stats_finalize_kernel<<<1, 256, 0, stream>>>(partial, g1, e1, scalebuf, shiftbuf,
                                                   nblk, Cout, 1.0f / (float)M);
      norm_relu_kernel<<<4096, 256, 0, stream>>>(bufA, scalebuf, shiftbuf, (size_t)M * Cout, Cout);
    }
    maxpool_out_kernel<<<cdiv((long long)B_ * S_ * c2[i], 256), 256, 0, stream>>>(
        bufA, out_points, K, c2[i], choff[i]);
  }
}